// VdwOQDO_47098611368250
// MI455X (gfx1250) — compile-verified
//
#include <hip/hip_runtime.h>
#include <cmath>

// ---------------------------------------------------------------------------
// vdW (OQDO) per-atom energy: fused edge pipeline for gfx1250 (MI455X).
// out[src] += 0.5 * switch * (exij - epair)   (fuses edisp + ex segment sums)
// ---------------------------------------------------------------------------

typedef int   v4i __attribute__((ext_vector_type(4)));
typedef float v4f __attribute__((ext_vector_type(4)));
typedef float v2f __attribute__((ext_vector_type(2)));

#define DEVFN static __device__ __forceinline__

DEVFN float frcp (float x) { return __builtin_amdgcn_rcpf(x); }          // v_rcp_f32
DEVFN float fexp2(float x) { return __builtin_amdgcn_exp2f(x); }          // v_exp_f32
DEVFN float flog2(float x) { return __builtin_amdgcn_logf(x); }           // v_log_f32
DEVFN float fexpn(float x) { return fexp2(x * 1.44269504088896341f); }    // e^x

// Full per-pair OQDO math (damped dispersion + exchange), f32 throughout.
// Returns 0.5*(exij - epair); caller multiplies by switch and scatters.
DEVFN float pair_contrib(float c6i, float c6j, float ai, float aj,
                         float rij, float Kconst)
{
    float alphaij = 0.5f * (ai + aj);
    float c6ij = 2.0f * ai * aj * c6i * c6j * frcp(fmaf(c6i, aj * aj, c6j * ai * ai));

    // Re = (alphaij * 128/FSC^(4/3))^(1/7)  via exp2(log2(x)/7)
    float Re  = fexp2(flog2(alphaij * Kconst) * (1.0f / 7.0f));
    float Re2 = Re * Re;
    float Re4 = Re2 * Re2;

    float num = fmaf(-7.21940151e-07f, Re4, fmaf(0.00127066988f, Re2,
                fmaf(-0.0376191669f, Re, 0.483053463f)));
    float den = fmaf(0.023741089f, Re2, fmaf(-0.0316915319f, Re, 0.038421212f));
    float muw = num * frcp(den);

    float inv_muw = frcp(muw);
    float c8ij  = 5.0f    * c6ij * inv_muw;
    float c10ij = 30.625f * c6ij * inv_muw * inv_muw;   // 245/8

    // ---- dispersion damping at actual rij ----
    float r2  = rij * rij;
    float z   = 0.5f * muw * r2;
    float ez  = fexpn(-z);
    float z2  = z * z;
    float f6  = 1.0f - ez * (1.0f + z + 0.5f * z2 + (1.0f / 6.0f) * z * z2);
    float f8  = f6 - (1.0f / 24.0f)  * ez * z2 * z2;
    float f10 = f8 - (1.0f / 120.0f) * ez * z2 * z2 * z;

    float inv_r2  = frcp(r2);
    float inv_r6  = inv_r2 * inv_r2 * inv_r2;
    float inv_r8  = inv_r6 * inv_r2;
    float inv_r10 = inv_r8 * inv_r2;
    float epair = f6 * c6ij * inv_r6 + f8 * c8ij * inv_r8 + f10 * c10ij * inv_r10;

    // ---- exchange part (damped, evaluated at Re) ----
    float w  = (4.0f / 3.0f) * c6ij * frcp(alphaij * alphaij);
    float q2 = alphaij * muw * w;

    float ze  = 0.5f * muw * Re2;
    float eze = fexpn(-ze);
    float ze2 = ze * ze;
    float s6  = eze * (1.0f + ze + 0.5f * ze2 + (1.0f / 6.0f) * ze * ze2);
    float f6e = 1.0f - s6;

    float muwRe = muw * Re;
    float mR2   = muwRe * muwRe;
    float df6e = muwRe * s6
               - eze * (muwRe + 0.5f * Re * mR2 + 0.125f * Re2 * mR2 * muwRe);

    float s8   = (1.0f / 24.0f) * eze * ze2 * ze2;
    float f8e  = f6e - s8;
    float df8e = df6e + muwRe * s8 - (1.0f / 48.0f) * eze * Re2 * Re * mR2 * mR2;

    float s10   = (1.0f / 120.0f) * eze * ze2 * ze2 * ze;
    float f10e  = f8e - s10;
    float df10e = df8e + muwRe * s10
                - (1.0f / 384.0f) * eze * Re4 * mR2 * mR2 * muwRe;

    float denA     = 2.0f * c6ij * Re2 * (6.0f * f6e - Re * df6e);
    float inv_denA = frcp(denA);
    float A = 0.5f + c8ij  * (8.0f  * f8e  - Re * df8e ) * inv_denA
                   + c10ij * (10.0f * f10e - Re * df10e) * inv_denA * frcp(Re2);

    float inv_rij = rij * inv_r2;           // 1/r without an extra v_rcp
    float exij = A * q2 * ez * inv_rij;

    return 0.5f * (exij - epair);
}

DEVFN void do_edge(int s, int d, float dist_ang, float swv,
                   const int* __restrict__ species,
                   const float* __restrict__ c6t, const float* __restrict__ alt,
                   const v2f* __restrict__ npack,
                   bool pre, float* __restrict__ out, float Kconst)
{
    float rij = dist_ang * 1.8897261254578281f;   // Angstrom -> Bohr (1/ANG)
    float c6i, c6j, ai, aj;
    if (pre) {                                    // packed {c6, alpha}: 2x b64 gathers
        v2f pi = npack[s];
        v2f pj = npack[d];
        c6i = pi.x; ai = pi.y;
        c6j = pj.x; aj = pj.y;
    } else {                                      // fallback: double indirection
        int spi = species[s], spj = species[d];
        c6i = c6t[spi]; c6j = c6t[spj];
        ai  = alt[spi]; aj  = alt[spj];
    }
    float contrib = swv * pair_contrib(c6i, c6j, ai, aj, rij, Kconst);
    unsafeAtomicAdd(&out[s], contrib);            // global_atomic_add_f32, no return
}

template <bool PRE>
__global__ void __launch_bounds__(256)
vdw_edge_kernel(const int* __restrict__ esrc, const int* __restrict__ edst,
                const float* __restrict__ dist, const float* __restrict__ sw,
                const int* __restrict__ species,
                const float* __restrict__ c6t, const float* __restrict__ alt,
                const v2f* __restrict__ npack,
                float* __restrict__ out, int n_edges, float Kconst)
{
    const int nvec   = n_edges >> 2;
    const int tid    = blockIdx.x * blockDim.x + threadIdx.x;
    const int stride = gridDim.x * blockDim.x;

    const v4i* s4 = (const v4i*)esrc;
    const v4i* d4 = (const v4i*)edst;
    const v4f* r4 = (const v4f*)dist;
    const v4f* w4 = (const v4f*)sw;

    for (int i = tid; i < nvec; i += stride) {
        // Speculative prefetch of next grid-stride iteration (TH=0 is
        // speculative per ISA 7.3: invalid translations silently dropped,
        // so no range guard needed -> no extra branch).
        __builtin_prefetch(&s4[i + stride], 0, 0);
        __builtin_prefetch(&d4[i + stride], 0, 0);
        __builtin_prefetch(&r4[i + stride], 0, 0);
        __builtin_prefetch(&w4[i + stride], 0, 0);

        // 128-bit non-temporal loads: streamed once, keep L2 for gather tables
        v4i s = __builtin_nontemporal_load(&s4[i]);
        v4i d = __builtin_nontemporal_load(&d4[i]);
        v4f r = __builtin_nontemporal_load(&r4[i]);
        v4f w = __builtin_nontemporal_load(&w4[i]);
        #pragma unroll
        for (int k = 0; k < 4; ++k)
            do_edge(s[k], d[k], r[k], w[k], species, c6t, alt, npack,
                    PRE, out, Kconst);
    }

    // Tail (n_edges % 4): negligible, handled by one thread
    if (tid == 0) {
        for (int e = nvec << 2; e < n_edges; ++e)
            do_edge(esrc[e], edst[e], dist[e], sw[e], species, c6t, alt,
                    npack, PRE, out, Kconst);
    }
}

// Force the hot kernels to be emitted first so the disasm snippet shows them.
template __global__ void vdw_edge_kernel<true >(const int*, const int*,
    const float*, const float*, const int*, const float*, const float*,
    const v2f*, float*, int, float);
template __global__ void vdw_edge_kernel<false>(const int*, const int*,
    const float*, const float*, const int*, const float*, const float*,
    const v2f*, float*, int, float);

// Fused prep: zero the per-node accumulator and (optionally) collapse
// table[species[i]] into a packed per-node {c6, alpha} array.
__global__ void vdw_prep_kernel(const int* __restrict__ species,
                                const float* __restrict__ c6t,
                                const float* __restrict__ alt,
                                v2f* __restrict__ npack,
                                float* __restrict__ out,
                                int n_nodes, int n_out, bool pre)
{
    int i = blockIdx.x * blockDim.x + threadIdx.x;
    if (i < n_out) out[i] = 0.0f;
    if (pre && i < n_nodes) {
        int sp = species[i];
        v2f p; p.x = c6t[sp]; p.y = alt[sp];
        npack[i] = p;
    }
}

extern "C" void kernel_launch(void* const* d_in, const int* in_sizes, int n_in,
                              void* d_out, int out_size, void* d_ws, size_t ws_size,
                              hipStream_t stream)
{
    // setup_inputs() order:
    const int*   species  = (const int*)  d_in[0];
    const int*   edge_src = (const int*)  d_in[1];
    const int*   edge_dst = (const int*)  d_in[2];
    const float* dist     = (const float*)d_in[3];
    const float* swv      = (const float*)d_in[4];
    const float* c6t      = (const float*)d_in[5];
    const float* alt      = (const float*)d_in[6];
    float*       out      = (float*)d_out;

    const int n_nodes = in_sizes[0];
    const int n_edges = in_sizes[1];

    // K = 128 / FSC^(4/3), computed in double on host (matches JAX python-float const)
    const double FSC = 7.2973525693e-3;
    const float  Kconst = (float)(128.0 / std::pow(FSC, 4.0 / 3.0));

    const size_t need = (size_t)n_nodes * sizeof(float) * 2;
    const bool   pre  = (ws_size >= need) && (d_ws != nullptr);
    v2f* npack = (v2f*)d_ws;

    // 1) zero output + build packed per-node table (one launch)
    {
        int n = (out_size > n_nodes) ? out_size : n_nodes;
        dim3 g((n + 255) / 256), b(256);
        vdw_prep_kernel<<<g, b, 0, stream>>>(species, c6t, alt, npack, out,
                                             n_nodes, out_size, pre);
    }

    // 2) fused edge pipeline: ~1 vec4 per thread, grid capped at 1024 blocks
    {
        int nvec   = n_edges >> 2;
        int blocks = (nvec + 255) / 256;
        if (blocks > 1024) blocks = 1024;
        if (blocks < 1)    blocks = 1;
        dim3 g(blocks), b(256);
        if (pre)
            vdw_edge_kernel<true ><<<g, b, 0, stream>>>(edge_src, edge_dst, dist, swv,
                species, c6t, alt, npack, out, n_edges, Kconst);
        else
            vdw_edge_kernel<false><<<g, b, 0, stream>>>(edge_src, edge_dst, dist, swv,
                species, c6t, alt, npack, out, n_edges, Kconst);
    }
}